// RegionLayer_deprecated_9500467658863
// MI455X (gfx1250) — compile-verified
//
#include <hip/hip_runtime.h>

// Problem constants (from reference)
#define G_B   16
#define G_A   5
#define G_C   80
#define G_G   32
#define G_NT  256
#define N_CELLS (G_B * G_A * G_G * G_G)   // 81920
#define CH    (G_C + 5)                   // 85
#define EPSI  1e-16f
#define THRESH 0.6f
#define SEEN_WARMUP 12800

// Workspace layout (as u32 words): [0..7] float accumulators, [8..2567] obj bitmask
// acc indices: 0=Scoord 1=Sobj 2=Scls 3=Snoobj 4=CntNoobj 5=CntObj
#define WS_WORDS 2568

__device__ __constant__ float c_aw[5] = {1.3221f, 3.19275f, 5.05587f, 9.47112f, 11.2364f};
__device__ __constant__ float c_ah[5] = {1.73145f, 4.00944f, 8.09892f, 4.84053f, 10.0071f};

typedef __attribute__((ext_vector_type(2))) float v2f;
typedef __attribute__((ext_vector_type(8))) float v8f;

__device__ __forceinline__ float sigm(float v) { return 1.0f / (1.0f + __expf(-v)); }

// Wave32 sum reduction through the matrix pipe.
// A(16x4) carries one value per lane in the K0/K2 slots (A-VGPR0), zeros in K1/K3.
// With B = ones(4x16): D[m,n] = v[m] + v[m+16] for every n (layout-independent in B).
// D layout: VGPR j -> lanes 0-15 = row j, lanes 16-31 = row j+8, so summing the
// 8 D VGPRs gives each half-wave partial; one shfl_xor(16) completes the sum.
__device__ __forceinline__ float wave_sum(float v) {
  v2f a; a.x = v;    a.y = 0.0f;
  v2f b; b.x = 1.0f; b.y = 1.0f;
  v8f c = {};
  c = __builtin_amdgcn_wmma_f32_16x16x4_f32(false, a, false, b, (short)0, c, false, false);
  float s = c[0] + c[1] + c[2] + c[3] + c[4] + c[5] + c[6] + c[7];
  s += __shfl_xor(s, 16, 32);
  return s;
}

__device__ __forceinline__ float iou_xywh(float ax, float ay, float aw, float ah,
                                          float bx, float by, float bw, float bh) {
  float ax1 = ax - aw * 0.5f, ax2 = ax + aw * 0.5f;
  float ay1 = ay - ah * 0.5f, ay2 = ay + ah * 0.5f;
  float bx1 = bx - bw * 0.5f, bx2 = bx + bw * 0.5f;
  float by1 = by - bh * 0.5f, by2 = by + bh * 0.5f;
  float iw = fmaxf(fminf(ax2, bx2) - fmaxf(ax1, bx1), 0.0f);
  float ih = fmaxf(fminf(ay2, by2) - fmaxf(ay1, by1), 0.0f);
  float inter = iw * ih;
  float uni = (ax2 - ax1) * (ay2 - ay1) + (bx2 - bx1) * (by2 - by1) - inter + EPSI;
  return inter / uni;
}

__global__ void k_zero(unsigned* __restrict__ ws) {
  int i = blockIdx.x * 256 + threadIdx.x;
  if (i < WS_WORDS) ws[i] = 0u;
}

// One block of 256 threads: one thread per target.
// Builds the obj bitmask and accumulates all obj-cell loss terms.
__global__ void k_targets(const float* __restrict__ x, const float* __restrict__ tg,
                          float* __restrict__ acc, unsigned* __restrict__ objmask) {
  __shared__ int s_cell[G_NT];
  __shared__ int s_lbl[G_NT];
  int t = threadIdx.x;

  float gx = tg[6 * t + 2] * (float)G_G;
  float gy = tg[6 * t + 3] * (float)G_G;
  float gw = tg[6 * t + 4] * (float)G_G;
  float gh = tg[6 * t + 5] * (float)G_G;
  int b   = min(max((int)tg[6 * t + 0], 0), G_B - 1);
  int lbl = min(max((int)tg[6 * t + 1], 0), G_C - 1);

  // best anchor (first-max semantics like jnp.argmax)
  int best = 0; float bestr = -1.0f;
#pragma unroll
  for (int a = 0; a < G_A; a++) {
    float iw = fminf(c_aw[a], gw), ih = fminf(c_ah[a], gh);
    float inter = iw * ih;
    float uni = c_aw[a] * c_ah[a] + gw * gh - inter + EPSI;
    float r = inter / uni;
    if (r > bestr) { bestr = r; best = a; }
  }
  int gi = min(max((int)gx, 0), G_G - 1);
  int gj = min(max((int)gy, 0), G_G - 1);
  int cell = ((b * G_A + best) * G_G + gj) * G_G + gi;

  s_cell[t] = cell;
  s_lbl[t]  = lbl;
  atomicOr(&objmask[cell >> 5], 1u << (cell & 31));
  __syncthreads();

  // last-scatter-wins: winner = no later target maps to same cell
  bool winner = true;
  for (int u = t + 1; u < G_NT; u++) if (s_cell[u] == cell) winner = false;

  float scoord = 0.0f, sobj = 0.0f, scls = 0.0f, cobj = 0.0f;
  if (winner) {
    // tcls is a union over *all* targets at this cell (per-label elements)
    unsigned lm0 = 0, lm1 = 0, lm2 = 0;
    for (int u = 0; u < G_NT; u++) {
      if (s_cell[u] == cell) {
        int L = s_lbl[u];
        if (L < 32) lm0 |= 1u << L;
        else if (L < 64) lm1 |= 1u << (L - 32);
        else lm2 |= 1u << (L - 64);
      }
    }

    const float* p = x + (size_t)(b * (G_A * CH) + best * CH) * (G_G * G_G) + (gj * G_G + gi);
    float px = sigm(p[0]);
    float py = sigm(p[1 * 1024]);
    float pw = p[2 * 1024];
    float ph = p[3 * 1024];
    float pconf = sigm(p[4 * 1024]);

    float aw = c_aw[best], ah = c_ah[best];
    float bx = px + (float)gi, by = py + (float)gj;
    float bw = __expf(pw) * aw, bh = __expf(ph) * ah;

    float iou = iou_xywh(bx, by, bw, bh, gx, gy, gw, gh);
    float dc = pconf - iou;
    sobj = dc * dc;

    float cm = 2.0f - tg[6 * t + 4] * tg[6 * t + 5];   // COORD_SCALE = 1
    float tx = gx - floorf(gx), ty = gy - floorf(gy);
    float tw = __logf(gw / aw), th = __logf(gh / ah);
    float dx = px - tx, dy = py - ty, dw = pw - tw, dh = ph - th;
    scoord = cm * (dx * dx + dy * dy + dw * dw + dh * dh);
    cobj = 1.0f;

    for (int c = 0; c < G_C; c++) {
      float pc = sigm(p[(5 + c) * 1024]);
      bool pos = (c < 32) ? ((lm0 >> c) & 1)
               : (c < 64) ? ((lm1 >> (c - 32)) & 1)
                          : ((lm2 >> (c - 64)) & 1);
      float lg = pos ? __logf(pc) : __logf(1.0f - pc);
      scls += -fmaxf(lg, -100.0f);
    }
  }

  // reconverged: EXEC all-ones -> WMMA reductions legal
  scoord = wave_sum(scoord);
  sobj   = wave_sum(sobj);
  scls   = wave_sum(scls);
  cobj   = wave_sum(cobj);
  if ((threadIdx.x & 31) == 0) {
    atomicAdd(&acc[0], scoord);
    atomicAdd(&acc[1], sobj);
    atomicAdd(&acc[2], scls);
    atomicAdd(&acc[5], cobj);
  }
}

// 320 blocks x 256 threads: one thread per (b,a,gj,gi) cell.
// Reads only 5 of 85 channels; target boxes live in LDS as float4 so the hot
// loop is one ds_load_b128 + one ds_load_b32 (uniform-address broadcast) per pair.
__global__ void k_noobj(const float* __restrict__ x, const float* __restrict__ tg,
                        const int* __restrict__ seen,
                        float* __restrict__ acc, const unsigned* __restrict__ objmask) {
  __shared__ float4 s_box[G_NT];   // {x1, y1, x2, y2}
  __shared__ float  s_rhs[G_NT];   // THRESH * area2
  int tid = threadIdx.x;
  {
    float gx = tg[6 * tid + 2] * (float)G_G;
    float gy = tg[6 * tid + 3] * (float)G_G;
    float gw = tg[6 * tid + 4] * (float)G_G;
    float gh = tg[6 * tid + 5] * (float)G_G;
    s_box[tid] = make_float4(gx - gw * 0.5f, gy - gh * 0.5f,
                             gx + gw * 0.5f, gy + gh * 0.5f);
    s_rhs[tid] = THRESH * (gw * gh);
  }
  __syncthreads();

  int cell = blockIdx.x * 256 + tid;
  int b  = cell / (G_A * 1024);
  int r  = cell - b * (G_A * 1024);
  int a  = r >> 10;
  int r2 = r & 1023;
  int gj = r2 >> 5, gi = r2 & 31;

  const float* p = x + (size_t)(b * (G_A * CH) + a * CH) * 1024 + r2;
  float px = sigm(p[0]);
  float py = sigm(p[1 * 1024]);
  float pw = p[2 * 1024];
  float ph = p[3 * 1024];
  float pconf = sigm(p[4 * 1024]);

  float bw = __expf(pw) * c_aw[a], bh = __expf(ph) * c_ah[a];
  float bx = px + (float)gi, by = py + (float)gj;
  float x1 = bx - bw * 0.5f, x2 = bx + bw * 0.5f;
  float y1 = by - bh * 0.5f, y2 = by + bh * 0.5f;
  float a1 = bw * bh;

  // exists t: iou > 0.6  <=>  max_t [1.6*inter_t - 0.6*a2_t] > 0.6*(a1+eps)
  float m = -1e30f;
#pragma unroll 4
  for (int t = 0; t < G_NT; t++) {
    float4 bb = s_box[t];
    float iw = fmaxf(fminf(x2, bb.z) - fmaxf(x1, bb.x), 0.0f);
    float ih = fmaxf(fminf(y2, bb.w) - fmaxf(y1, bb.y), 0.0f);
    float inter = iw * ih;
    m = fmaxf(m, (1.0f + THRESH) * inter - s_rhs[t]);
  }
  bool hit = m > THRESH * (a1 + EPSI);
  bool obj = (objmask[cell >> 5] >> (cell & 31)) & 1u;

  bool noobj = (!obj) && (!hit);
  float sn = noobj ? pconf * pconf : 0.0f;   // tconf == 0 off target cells
  float cn = noobj ? 1.0f : 0.0f;

  sn = wave_sum(sn);
  cn = wave_sum(cn);
  if ((threadIdx.x & 31) == 0) {
    atomicAdd(&acc[3], sn);
    atomicAdd(&acc[4], cn);
  }

  if (*seen < SEEN_WARMUP) {   // uniform branch: EXEC all-ones inside -> WMMA legal
    // warm-up base: coord_mask=1, tx=ty=0.5, tw=th=0 at non-target cells
    float dx = px - 0.5f, dy = py - 0.5f;
    float sc = obj ? 0.0f : (dx * dx + dy * dy + pw * pw + ph * ph);
    sc = wave_sum(sc);
    if ((threadIdx.x & 31) == 0) atomicAdd(&acc[0], sc);
  }
}

__global__ void k_final(const float* __restrict__ acc, float* __restrict__ out) {
  float nobj = acc[5], nnoobj = acc[4];
  float loss = acc[0] * (1.0f / (float)N_CELLS)              // loss_coord (4 mse's share the mean denom)
             + (nnoobj / nobj) * acc[1] / nobj               // loss_conf_obj
             + acc[3] / nnoobj                               // loss_conf_noobj (NOOBJ_SCALE=1)
             + acc[2] / (nobj * (float)G_C);                 // loss_cls (CLASS_SCALE=1)
  out[0] = loss;
}

extern "C" void kernel_launch(void* const* d_in, const int* in_sizes, int n_in,
                              void* d_out, int out_size, void* d_ws, size_t ws_size,
                              hipStream_t stream) {
  const float* x    = (const float*)d_in[0];
  const float* tg   = (const float*)d_in[1];
  const int*   seen = (const int*)d_in[2];
  float*    out  = (float*)d_out;
  float*    acc  = (float*)d_ws;
  unsigned* mask = ((unsigned*)d_ws) + 8;

  k_zero<<<(WS_WORDS + 255) / 256, 256, 0, stream>>>((unsigned*)d_ws);
  k_targets<<<1, 256, 0, stream>>>(x, tg, acc, mask);
  k_noobj<<<N_CELLS / 256, 256, 0, stream>>>(x, tg, seen, acc, mask);
  k_final<<<1, 1, 0, stream>>>(acc, out);
}